// Model_32186484916749
// MI455X (gfx1250) — compile-verified
//
#include <hip/hip_runtime.h>
#include <math.h>

// Problem constants (fixed by the reference)
#define NN 128
#define CC 64
#define TT 256
#define VV 25
#define DD 64
#define VC 1600              // V*C == V*D
#define TVROWS 6400          // T*V rows per sample
#define ROWS_PER_BLK 128     // 8 waves x 16-row M tiles
#define RAW_T 7              // max t-slices a 128-row chunk can span (ceil(128/25)+1)

typedef float v2f __attribute__((ext_vector_type(2)));
typedef float v8f __attribute__((ext_vector_type(8)));

// ---------------------------------------------------------------------------
// Kernel 1: global average pool over (T,V) per (n,c).  One block per (n,c).
// ---------------------------------------------------------------------------
__global__ __launch_bounds__(256) void pool_kernel(const float* __restrict__ x0,
                                                   float* __restrict__ pooled) {
  __shared__ float red[256];
  const int b = blockIdx.x;                  // n*64 + c
  const float* p = x0 + (size_t)b * TVROWS;  // contiguous 6400 floats
  float s = 0.f;
  for (int i = threadIdx.x; i < TVROWS; i += 256) s += p[i];
  red[threadIdx.x] = s;
  __syncthreads();
  for (int off = 128; off > 0; off >>= 1) {
    if (threadIdx.x < off) red[threadIdx.x] += red[threadIdx.x + off];
    __syncthreads();
  }
  if (threadIdx.x == 0) pooled[b] = red[0] * (1.f / (float)TVROWS);
}

// ---------------------------------------------------------------------------
// Kernel 2: SE MLP + softmax gate, fused weights Wf[n,c,d] and bias bf[n,d].
// One block (128 threads) per sample n.
// ---------------------------------------------------------------------------
__global__ __launch_bounds__(128) void se_kernel(
    const float* __restrict__ pooled, const int* __restrict__ epoch,
    const float* __restrict__ fc1_w, const float* __restrict__ fc1_b,
    const float* __restrict__ fc2_w, const float* __restrict__ fc2_b,
    const float* __restrict__ W, const float* __restrict__ bb,
    float* __restrict__ Wf, float* __restrict__ bf) {
  __shared__ float pl[64];
  __shared__ float h[16];
  __shared__ float gate[4];
  const int n = blockIdx.x, tid = threadIdx.x;
  if (tid < 64) pl[tid] = pooled[n * 64 + tid];
  __syncthreads();
  if (tid < 16) {
    float s = fc1_b[tid];
#pragma unroll
    for (int c = 0; c < 64; ++c) s += pl[c] * fc1_w[tid * 64 + c];
    h[tid] = s > 0.f ? s : 0.f;
  }
  __syncthreads();
  if (tid < 4) {
    float s = fc2_b[tid];
#pragma unroll
    for (int r = 0; r < 16; ++r) s += h[r] * fc2_w[tid * 16 + r];
    gate[tid] = s;
  }
  __syncthreads();
  if (tid == 0) {
    const int ep = epoch[0];
    const float tao = (ep >= 60) ? 1.f : (-(30.f - 1.f) / 60.f) * (float)ep + 30.f;
    const float it = 1.f / tao;
    float l0 = gate[0] * it, l1 = gate[1] * it, l2 = gate[2] * it, l3 = gate[3] * it;
    float m = fmaxf(fmaxf(l0, l1), fmaxf(l2, l3));
    float e0 = expf(l0 - m), e1 = expf(l1 - m), e2 = expf(l2 - m), e3 = expf(l3 - m);
    float inv = 1.f / (e0 + e1 + e2 + e3);
    gate[0] = e0 * inv; gate[1] = e1 * inv; gate[2] = e2 * inv; gate[3] = e3 * inv;
  }
  __syncthreads();
  const float g0 = gate[0], g1 = gate[1], g2 = gate[2], g3 = gate[3];
  for (int i = tid; i < 4096; i += 128)
    Wf[n * 4096 + i] = g0 * W[i] + g1 * W[4096 + i] + g2 * W[8192 + i] + g3 * W[12288 + i];
  if (tid < 64)
    bf[n * 64 + tid] = g0 * bb[tid] + g1 * bb[64 + tid] + g2 * bb[128 + tid] + g3 * bb[192 + tid];
}

// ---------------------------------------------------------------------------
// Kernel 3: precompute tanh(mask)+1 and zero the BN stat accumulators.
// ---------------------------------------------------------------------------
__global__ void prep_kernel(const float* __restrict__ mask, float* __restrict__ tm,
                            float* __restrict__ gSum, float* __restrict__ gSsq) {
  const int i = blockIdx.x * blockDim.x + threadIdx.x;
  if (i < VC) {
    tm[i] = tanhf(mask[i]) + 1.f;
    gSum[i] = 0.f;
    gSsq[i] = 0.f;
  }
}

// ---------------------------------------------------------------------------
// Kernel 5 (between the two GEMM passes): finalize BN -> per-(pre-gather)
// feature scale/shift, plus the inverse of the shift_out permutation.
// ---------------------------------------------------------------------------
__global__ void finalize_kernel(const int* __restrict__ shift_out,
                                const float* __restrict__ gSum,
                                const float* __restrict__ gSsq,
                                const float* __restrict__ gamma,
                                const float* __restrict__ beta,
                                float* __restrict__ scaleA, float* __restrict__ shiftA,
                                int* __restrict__ invA) {
  const int j = blockIdx.x * blockDim.x + threadIdx.x;
  if (j < VC) {
    const int f = shift_out[j];
    const float R = (float)(NN * TT);
    const float mu = gSum[f] / R;
    const float var = gSsq[f] / R - mu * mu;
    const float sc = rsqrtf(var + 1e-5f) * gamma[j];
    scaleA[f] = sc;
    shiftA[f] = beta[j] - mu * sc;
    invA[f] = j;
  }
}

// ---------------------------------------------------------------------------
// Kernels 4 & 6: fused shift-in gather + mask + per-sample GEMM (fp32 WMMA)
//   STATS=true : accumulate per-feature sum / sum-of-squares (BN pass 1)
//   STATS=false: apply BN + shift-out scatter + residual + ReLU   (BN pass 2)
// Grid: (50 row-chunks, 128 samples); block = 256 threads = 8 waves,
// each wave owns a 16-row x 64-col output strip via V_WMMA_F32_16X16X4_F32.
// ---------------------------------------------------------------------------
template <bool STATS>
__global__ __launch_bounds__(256) void fused_gemm_kernel(
    const float* __restrict__ x0, const float* __restrict__ Wf,
    const float* __restrict__ bf, const float* __restrict__ tm,
    const int* __restrict__ shift_in,
    float* __restrict__ gSum, float* __restrict__ gSsq,
    const float* __restrict__ scaleA, const float* __restrict__ shiftA,
    const int* __restrict__ invA, float* __restrict__ out) {
  __shared__ float WfL[64 * 64];                 // B matrix  [K=64][N=64]   16 KB
  __shared__ float xm[ROWS_PER_BLK * 65];        // A tile, padded rows      33 KB
  __shared__ float raw[RAW_T * VC];              // staged x0 slices         45 KB
  __shared__ float aux[2 * VC];                  // stats OR scale/shift     13 KB
  __shared__ int auxI[VC];                       // inverse shift_out         6 KB

  const int tid = threadIdx.x;
  const int n = blockIdx.y;
  const int rows0 = blockIdx.x * ROWS_PER_BLK;
  const int t_lo = rows0 / VV;
  const int t_hi = (rows0 + ROWS_PER_BLK - 1) / VV;
  const int t_cnt = t_hi - t_lo + 1;

  // ---- stage B (Wf[n]) ----
  for (int i = tid; i < 4096; i += 256) WfL[i] = Wf[n * 4096 + i];
  // ---- stage raw x0 slices x0[n, :, t_lo..t_hi, :]  (coalesced-ish) ----
  for (int e = tid; e < t_cnt * VC; e += 256) {
    const int tt = e / VC;
    const int rem = e - tt * VC;
    const int c = rem / VV;
    const int v = rem - c * VV;
    raw[e] = x0[((size_t)(n * 64 + c)) * TVROWS + (t_lo + tt) * VV + v];
  }
  // ---- stage epilogue tables ----
  for (int i = tid; i < VC; i += 256) {
    if (STATS) {
      aux[i] = 0.f;
      aux[VC + i] = 0.f;
    } else {
      aux[i] = scaleA[i];
      aux[VC + i] = shiftA[i];
      auxI[i] = invA[i];
    }
  }
  __syncthreads();

  // ---- build permuted + masked A tile in LDS ----
  for (int e = tid; e < ROWS_PER_BLK * 64; e += 256) {
    const int rl = e >> 6;
    const int cc = e & 63;
    const int rg = rows0 + rl;
    const int t = rg / VV;
    const int v = rg - t * VV;
    const int s = shift_in[v * 64 + cc];     // permutation over [0,1600)
    const int vs = s >> 6;                   // source joint
    const int cs = s & 63;                   // source channel
    xm[rl * 65 + cc] = raw[(t - t_lo) * VC + cs * VV + vs] * tm[v * 64 + cc];
  }
  __syncthreads();

  // ---- WMMA GEMM: each wave -> 16 rows x 64 cols ----
  const int wave = tid >> 5;
  const int lane = tid & 31;
  const int laneM = lane & 15;
  const int hi = lane >> 4;                  // ISA: lanes 16-31 hold K+2,K+3
  const int rowBase = wave * 16;

  v8f acc[4];
#pragma unroll
  for (int dt = 0; dt < 4; ++dt)
#pragma unroll
    for (int r = 0; r < 8; ++r) acc[dt][r] = 0.f;

#pragma unroll
  for (int k0 = 0; k0 < 64; k0 += 4) {
    const int kk = k0 + hi * 2;
    const int arow = rowBase + laneM;
    v2f a;
    a.x = xm[arow * 65 + kk];
    a.y = xm[arow * 65 + kk + 1];
#pragma unroll
    for (int dt = 0; dt < 4; ++dt) {
      const int col = dt * 16 + laneM;
      v2f bfr;
      bfr.x = WfL[kk * 64 + col];
      bfr.y = WfL[(kk + 1) * 64 + col];
      acc[dt] = __builtin_amdgcn_wmma_f32_16x16x4_f32(
          /*neg_a=*/false, a, /*neg_b=*/false, bfr,
          /*c_mod=*/(short)0, acc[dt], /*reuse_a=*/false, /*reuse_b=*/false);
    }
  }

  // ---- epilogue ----
#pragma unroll
  for (int dt = 0; dt < 4; ++dt) {
    const int col = dt * 16 + laneM;
    const float bv = bf[n * 64 + col];
#pragma unroll
    for (int r = 0; r < 8; ++r) {
      const float val = acc[dt][r] + bv;
      const int rl = rowBase + r + hi * 8;   // ISA C/D layout: lanes>=16 -> M+8
      const int rg = rows0 + rl;
      const int t = rg / VV;
      const int v = rg - t * VV;
      const int f = v * 64 + col;            // pre-gather feature id
      if (STATS) {
        atomicAdd(&aux[f], val);             // ds_add_f32
        atomicAdd(&aux[VC + f], val * val);
      } else {
        const int j = auxI[f];               // post-gather position
        const int dd = j & 63;
        const int vv = j >> 6;
        const size_t oi = ((size_t)(n * 64 + dd)) * TVROWS + t * VV + vv;
        float o = val * aux[f] + aux[VC + f] + raw[(t - t_lo) * VC + dd * VV + vv];
        out[oi] = o > 0.f ? o : 0.f;
      }
    }
  }

  if (STATS) {
    __syncthreads();
    for (int i = tid; i < VC; i += 256) {
      unsafeAtomicAdd(&gSum[i], aux[i]);       // global_atomic_add_f32
      unsafeAtomicAdd(&gSsq[i], aux[VC + i]);
    }
  }
}

// ---------------------------------------------------------------------------
// Host-side launch sequence (graph-capture safe: stream-ordered only)
// ---------------------------------------------------------------------------
extern "C" void kernel_launch(void* const* d_in, const int* in_sizes, int n_in,
                              void* d_out, int out_size, void* d_ws, size_t ws_size,
                              hipStream_t stream) {
  const float* x0 = (const float*)d_in[0];
  const int* epoch = (const int*)d_in[1];
  const float* fc1_w = (const float*)d_in[2];
  const float* fc1_b = (const float*)d_in[3];
  const float* fc2_w = (const float*)d_in[4];
  const float* fc2_b = (const float*)d_in[5];
  const float* W = (const float*)d_in[6];
  const float* bb = (const float*)d_in[7];
  const float* mask = (const float*)d_in[8];
  const float* gamma = (const float*)d_in[9];
  const float* beta = (const float*)d_in[10];
  const int* shift_in = (const int*)d_in[11];
  const int* shift_out = (const int*)d_in[12];
  float* out = (float*)d_out;

  // workspace layout (floats)
  float* w = (float*)d_ws;
  float* pooled = w;                 // 8192
  float* Wf = w + 8192;              // 524288
  float* bf = w + 532480;            // 8192
  float* tm = w + 540672;            // 1600
  float* gSum = w + 542272;          // 1600
  float* gSsq = w + 543872;          // 1600
  float* scaleA = w + 545472;        // 1600
  float* shiftA = w + 547072;        // 1600
  int* invA = (int*)(w + 548672);    // 1600

  pool_kernel<<<NN * CC, 256, 0, stream>>>(x0, pooled);
  se_kernel<<<NN, 128, 0, stream>>>(pooled, epoch, fc1_w, fc1_b, fc2_w, fc2_b, W, bb, Wf, bf);
  prep_kernel<<<(VC + 255) / 256, 256, 0, stream>>>(mask, tm, gSum, gSsq);

  dim3 grid(TVROWS / ROWS_PER_BLK, NN);  // (50, 128)
  fused_gemm_kernel<true><<<grid, 256, 0, stream>>>(
      x0, Wf, bf, tm, shift_in, gSum, gSsq, scaleA, shiftA, invA, out);
  finalize_kernel<<<(VC + 255) / 256, 256, 0, stream>>>(
      shift_out, gSum, gSsq, gamma, beta, scaleA, shiftA, invA);
  fused_gemm_kernel<false><<<grid, 256, 0, stream>>>(
      x0, Wf, bf, tm, shift_in, gSum, gSsq, scaleA, shiftA, invA, out);
}